// DynamicSobelKernel_12756052869128
// MI455X (gfx1250) — compile-verified
//
#include <hip/hip_runtime.h>
#include <stdint.h>

// DynamicSobel: fused 4x depthwise 3x3 + sqrt(weighted sum of squares).
// x:(8,32,512,512) f32 -> out same shape.  Memory-bound (~536MB -> ~23us @
// 23.3TB/s); kernel keeps VALU/instruction count low enough to stay on the
// HBM roofline:
//  - 64x16 output tile / 256 threads; halo tile in LDS, row stride 72 words
//    so the 64-wide interior is 16B-aligned
//  - interior staged with CDNA5 global_load_async_to_lds_b128 (ASYNCcnt),
//    halo columns with _b32, zero-fill for conv padding
//  - each thread computes 4 horizontal outputs from one 3x6 window
//    (b32+b128+b32 LDS reads per row), column diff/sum factorization,
//    single b128 global store.

#define IW 512
#define IH 512
#define NPLANES (8 * 32)
#define TX 64
#define TY 16
#define ROWS   (TY + 2)          // 18 tile rows (y0-1 .. y0+16)
#define STRIDE 72                // words per tile row (288B, 16B multiple)
// word 3 = left halo (gx = x0-1), words 4..67 = interior (gx = x0..x0+63),
// word 68 = right halo (gx = x0+64)
#define NTHREADS 256

__global__ __launch_bounds__(NTHREADS) void sobel_edge_kernel(
    const float* __restrict__ x, float* __restrict__ out,
    const float* __restrict__ pA, const float* __restrict__ pB,
    const float* __restrict__ pC, const float* __restrict__ pD)
{
    __shared__ __align__(16) float tile[ROWS * STRIDE];

    const int tid   = threadIdx.x;
    const int x0    = blockIdx.x * TX;
    const int y0    = blockIdx.y * TY;
    const int plane = blockIdx.z;

    const float* src = x + (size_t)plane * (IW * IH);

    // low 32 bits of an LDS object's generic address = DS byte offset
    const uint32_t lds_base = (uint32_t)(uintptr_t)(&tile[0]);

    // uniform scalar weights (issue early, overlap with staging)
    const float A  = *pA;
    const float B  = *pB;
    const float Cw = *pC;
    const float D  = *pD;

    // ---- interior staging: 18 rows x 16 lanes x b128 (4 floats/lane) ----
    for (int idx = tid; idx < ROWS * 16; idx += NTHREADS) {
        const int r  = idx >> 4;         // tile row 0..17
        const int l  = idx & 15;         // 16B chunk within row
        const int gy = y0 - 1 + r;
        const int word = r * STRIDE + 4 + 4 * l;
        if ((unsigned)gy < (unsigned)IH) {
            const uint32_t laddr = lds_base + (uint32_t)(word * 4);
            const uint32_t voff  = (uint32_t)(gy * IW + x0 + 4 * l) * 4u;
            asm volatile("global_load_async_to_lds_b128 %0, %1, %2"
                         :: "v"(laddr), "v"(voff), "s"(src)
                         : "memory");
        } else {
            float4 zz{0.0f, 0.0f, 0.0f, 0.0f};
            *(float4*)(&tile[word]) = zz;       // zero padding row
        }
    }

    // ---- halo columns: 18 rows x {left,right} x b32 ----
    if (tid < 2 * ROWS) {
        const int right = (tid >= ROWS);
        const int r  = tid - (right ? ROWS : 0);
        const int gy = y0 - 1 + r;
        const int gx = right ? (x0 + TX) : (x0 - 1);
        const int word = r * STRIDE + (right ? 68 : 3);
        if ((unsigned)gy < (unsigned)IH && (unsigned)gx < (unsigned)IW) {
            const uint32_t laddr = lds_base + (uint32_t)(word * 4);
            const uint32_t voff  = (uint32_t)(gy * IW + gx) * 4u;
            asm volatile("global_load_async_to_lds_b32 %0, %1, %2"
                         :: "v"(laddr), "v"(voff), "s"(src)
                         : "memory");
        } else {
            tile[word] = 0.0f;                  // zero padding corner/edge
        }
    }

    asm volatile("s_wait_asynccnt 0" ::: "memory");
    __syncthreads();

    // ---- compute: each thread -> 4 horizontal outputs ----
    const int g = tid & 15;          // column group: output cols 4g..4g+3
    const int r = tid >> 4;          // output row 0..15
    const int cb = 4 * g;

    // window: tile rows r..r+2, words cb+3 .. cb+8  (a[rr][0..5])
    float a[3][6];
#pragma unroll
    for (int rr = 0; rr < 3; ++rr) {
        const float* p = &tile[(r + rr) * STRIDE + cb + 3];
        a[rr][0] = p[0];                         // b32
        const float4 m = *(const float4*)(p + 1); // b128 (16B aligned)
        a[rr][1] = m.x; a[rr][2] = m.y; a[rr][3] = m.z; a[rr][4] = m.w;
        a[rr][5] = p[5];                         // b32
    }

    float vd[6], vs[6];
#pragma unroll
    for (int c = 0; c < 6; ++c) {
        vd[c] = a[2][c] - a[0][c];                       // vertical diff
        vs[c] = a[0][c] + 2.0f * a[1][c] + a[2][c];      // vertical sum
    }

    float4 res;
    float* rf = (float*)&res;
#pragma unroll
    for (int j = 0; j < 4; ++j) {
        const float gxv  = vs[j + 2] - vs[j];                    // kx
        const float gyv  = vd[j] + 2.0f * vd[j + 1] + vd[j + 2]; // ky
        const float dz   = a[1][j + 2] - a[1][j];                // a12-a10
        const float t    = a[2][j + 2] - a[0][j];                // a22-a00
        const float w    = a[2][j]     - a[0][j + 2];            // a20-a02
        const float g45  = (dz + vd[j + 1]) + 2.0f * t;          // k45
        const float g135 = (vd[j + 1] - dz) + 2.0f * w;          // k135
        const float e2 = A * gxv * gxv + B * gyv * gyv
                       + Cw * g45 * g45 + D * g135 * g135;
        rf[j] = sqrtf(e2);
    }

    float* dst = out + (size_t)plane * (IW * IH)
                     + (size_t)(y0 + r) * IW + (x0 + cb);
    *(float4*)dst = res;                         // b128 store, 16B aligned
}

extern "C" void kernel_launch(void* const* d_in, const int* in_sizes, int n_in,
                              void* d_out, int out_size, void* d_ws, size_t ws_size,
                              hipStream_t stream) {
    // setup_inputs order: x, kx, ky, k45, k135, alpha, beta, gamma, delta
    const float* x     = (const float*)d_in[0];
    const float* alpha = (const float*)d_in[5];
    const float* beta  = (const float*)d_in[6];
    const float* gamma = (const float*)d_in[7];
    const float* delta = (const float*)d_in[8];
    float* out = (float*)d_out;

    dim3 grid(IW / TX, IH / TY, NPLANES);   // 8 x 32 x 256 blocks
    sobel_edge_kernel<<<grid, NTHREADS, 0, stream>>>(x, out, alpha, beta, gamma, delta);
}